// LossFunction_33603824124377
// MI455X (gfx1250) — compile-verified
//
#include <hip/hip_runtime.h>
#include <hip/hip_bf16.h>

typedef __attribute__((ext_vector_type(2))) float v2f;
typedef __attribute__((ext_vector_type(4))) float v4f;
typedef __attribute__((ext_vector_type(8))) float v8f;

// ---------------------------------------------------------------------------
// Wave32 shuffle reduction
// ---------------------------------------------------------------------------
__device__ __forceinline__ float wave_reduce_sum(float v) {
    v += __shfl_down(v, 16, 32);
    v += __shfl_down(v, 8, 32);
    v += __shfl_down(v, 4, 32);
    v += __shfl_down(v, 2, 32);
    v += __shfl_down(v, 1, 32);
    return v;
}

// Per-element fused loss accumulation.
// p = class_output, t = class_target (0.0 or 1.0), r = reg_output, g = reg_target
__device__ __forceinline__ void acc_elem(float p, float t, float r, float g,
                                         float w0, float w1,
                                         float& s_bce, float& s_sq, float& s_cnt) {
    const bool is1 = (t != 0.0f);
    // BCE with t in {0,1}: -log(p) if t==1 else -log(1-p)
    const float bce = -logf(is1 ? p : (1.0f - p));
    const float w   = is1 ? w1 : w0;
    s_bce = fmaf(w, bce, s_bce);
    // Masked squared error over t == 0
    const float d  = r - g;
    const float sq = d * d;
    s_sq  += is1 ? 0.0f : sq;
    s_cnt += is1 ? 0.0f : 1.0f;
}

// ---------------------------------------------------------------------------
// Kernel 1: streaming partial reduction. Non-temporal 128-bit loads on all
// four arrays (data is single-use and larger than L2 -> TH=NT is the right
// cache policy). Writes 3 partial sums per block into ws:
//   ws[0..nb-1]        = sum(w * bce)
//   ws[nb..2nb-1]      = sum(masked (r-g)^2)
//   ws[2nb..3nb-1]     = count(t == 0)
// ---------------------------------------------------------------------------
__global__ void __launch_bounds__(256)
loss_partial_kernel(const float* __restrict__ co,
                    const float* __restrict__ ro,
                    const float* __restrict__ ct,
                    const float* __restrict__ rt,
                    const float* __restrict__ cw,
                    float* __restrict__ ws,
                    int n, int n4, int nblocks) {
    const float w0 = cw[0];
    const float w1 = cw[1];

    const v4f* __restrict__ co4 = reinterpret_cast<const v4f*>(co);
    const v4f* __restrict__ ro4 = reinterpret_cast<const v4f*>(ro);
    const v4f* __restrict__ ct4 = reinterpret_cast<const v4f*>(ct);
    const v4f* __restrict__ rt4 = reinterpret_cast<const v4f*>(rt);

    float s_bce = 0.0f, s_sq = 0.0f, s_cnt = 0.0f;

    const int stride = gridDim.x * blockDim.x;
#pragma unroll 2
    for (int i = blockIdx.x * blockDim.x + threadIdx.x; i < n4; i += stride) {
        const v4f p = __builtin_nontemporal_load(&co4[i]);
        const v4f t = __builtin_nontemporal_load(&ct4[i]);
        const v4f r = __builtin_nontemporal_load(&ro4[i]);
        const v4f g = __builtin_nontemporal_load(&rt4[i]);
        acc_elem(p[0], t[0], r[0], g[0], w0, w1, s_bce, s_sq, s_cnt);
        acc_elem(p[1], t[1], r[1], g[1], w0, w1, s_bce, s_sq, s_cnt);
        acc_elem(p[2], t[2], r[2], g[2], w0, w1, s_bce, s_sq, s_cnt);
        acc_elem(p[3], t[3], r[3], g[3], w0, w1, s_bce, s_sq, s_cnt);
    }
    // Scalar tail (N % 4), handled by one thread.
    if (blockIdx.x == 0 && threadIdx.x == 0) {
        for (int i = n4 * 4; i < n; ++i)
            acc_elem(co[i], ct[i], ro[i], rt[i], w0, w1, s_bce, s_sq, s_cnt);
    }

    // Wave32 reduction, then LDS reduction across the 8 waves.
    s_bce = wave_reduce_sum(s_bce);
    s_sq  = wave_reduce_sum(s_sq);
    s_cnt = wave_reduce_sum(s_cnt);

    __shared__ float part[8][3];
    const int lane = threadIdx.x & 31;
    const int wid  = threadIdx.x >> 5;
    if (lane == 0) {
        part[wid][0] = s_bce;
        part[wid][1] = s_sq;
        part[wid][2] = s_cnt;
    }
    __syncthreads();
    if (threadIdx.x == 0) {
        float a = 0.0f, b = 0.0f, c = 0.0f;
        const int nw = blockDim.x >> 5;
        for (int w = 0; w < nw; ++w) {
            a += part[w][0];
            b += part[w][1];
            c += part[w][2];
        }
        ws[blockIdx.x]               = a;
        ws[nblocks + blockIdx.x]     = b;
        ws[2 * nblocks + blockIdx.x] = c;
    }
}

// ---------------------------------------------------------------------------
// Kernel 2: final reduction of block partials via V_WMMA_F32_16X16X4_F32.
// One full wave, fully convergent (WMMA requires EXEC all-ones).
// A (16x4 f32) layout: lane l<16 -> VGPR0=A[l][0], VGPR1=A[l][1];
//                      lane l>=16 -> VGPR0=A[l-16][2], VGPR1=A[l-16][3].
// So lane l loading values (2l, 2l+1) covers all 64 A slots exactly once.
// With B = ones(4x16): D[m][n] = rowsum[m] for every n; accumulate tiles in C.
// Total = sum over rows = (sum of 8 C regs on lane 0) + (same on lane 16).
// ---------------------------------------------------------------------------
__device__ __forceinline__ float load_clamped(const float* __restrict__ p,
                                              int idx, int n) {
    // Branchless OOB handling: clamp the address (always in-bounds load),
    // select 0 for out-of-range. Keeps the wave convergent (no exec ladders).
    const int   c = (idx < n) ? idx : (n - 1);
    const float v = p[c];
    return (idx < n) ? v : 0.0f;
}

__global__ void __launch_bounds__(32)
loss_final_kernel(const float* __restrict__ ws,
                  float* __restrict__ out,
                  int nblocks, float inv_n) {
    const int lane  = threadIdx.x;         // 0..31, full wave
    const int tiles = (nblocks + 63) >> 6; // 64 values per WMMA tile

    v8f acc_bce = {};
    v8f acc_sq  = {};
    v8f acc_cnt = {};
    v2f ones;
    ones[0] = 1.0f;
    ones[1] = 1.0f;

    const float* __restrict__ wb = ws;
    const float* __restrict__ wq = ws + nblocks;
    const float* __restrict__ wc = ws + 2 * nblocks;

    for (int t = 0; t < tiles; ++t) {
        const int base = t * 64 + lane * 2;
        v2f a0, a1, a2;
        a0[0] = load_clamped(wb, base,     nblocks);
        a0[1] = load_clamped(wb, base + 1, nblocks);
        a1[0] = load_clamped(wq, base,     nblocks);
        a1[1] = load_clamped(wq, base + 1, nblocks);
        a2[0] = load_clamped(wc, base,     nblocks);
        a2[1] = load_clamped(wc, base + 1, nblocks);

        acc_bce = __builtin_amdgcn_wmma_f32_16x16x4_f32(
            false, a0, false, ones, (short)0, acc_bce, false, false);
        acc_sq = __builtin_amdgcn_wmma_f32_16x16x4_f32(
            false, a1, false, ones, (short)0, acc_sq, false, false);
        acc_cnt = __builtin_amdgcn_wmma_f32_16x16x4_f32(
            false, a2, false, ones, (short)0, acc_cnt, false, false);
    }

    // Per-lane: sum of rows 0..7 (lanes 0-15) or rows 8..15 (lanes 16-31).
    float sb = 0.0f, sq = 0.0f, sc = 0.0f;
#pragma unroll
    for (int r = 0; r < 8; ++r) {
        sb += acc_bce[r];
        sq += acc_sq[r];
        sc += acc_cnt[r];
    }
    const float tot_bce = __shfl(sb, 0, 32) + __shfl(sb, 16, 32);
    const float tot_sq  = __shfl(sq, 0, 32) + __shfl(sq, 16, 32);
    const float tot_cnt = __shfl(sc, 0, 32) + __shfl(sc, 16, 32);

    if (lane == 0) {
        const float class_loss = tot_bce * inv_n;
        const float reg_loss   = (tot_cnt > 0.0f) ? (tot_sq / tot_cnt) : 0.0f;
        out[0] = 0.5f * class_loss + 0.5f * reg_loss;
    }
}

// ---------------------------------------------------------------------------
// Host-side launcher
// ---------------------------------------------------------------------------
extern "C" void kernel_launch(void* const* d_in, const int* in_sizes, int n_in,
                              void* d_out, int out_size, void* d_ws, size_t ws_size,
                              hipStream_t stream) {
    const float* co = (const float*)d_in[0];  // class_output [N]
    const float* ro = (const float*)d_in[1];  // reg_output   [N]
    const float* ct = (const float*)d_in[2];  // class_target [N]
    const float* rt = (const float*)d_in[3];  // reg_target   [N]
    const float* cw = (const float*)d_in[4];  // class_weights [2]

    const int n  = in_sizes[0];
    const int n4 = n >> 2;

    int nblocks = 1024;
    while (nblocks > 1 && (size_t)(3 * nblocks) * sizeof(float) > ws_size)
        nblocks >>= 1;

    float* ws = (float*)d_ws;

    loss_partial_kernel<<<nblocks, 256, 0, stream>>>(co, ro, ct, rt, cw, ws,
                                                     n, n4, nblocks);
    loss_final_kernel<<<1, 32, 0, stream>>>(ws, (float*)d_out, nblocks,
                                            1.0f / (float)n);
}